// SGCNModel_77601469104425
// MI455X (gfx1250) — compile-verified
//
#include <hip/hip_runtime.h>
#include <hip/hip_bf16.h>

#define NB   8
#define NN   256
#define HH   128
#define LL   3
#define NF   6
#define EF   15
#define TD   4
#define FHD  64

typedef __attribute__((ext_vector_type(2))) float v2f;
typedef __attribute__((ext_vector_type(8))) float v8f;

// ---------------------------------------------------------------------------
// mask (float) + degree from adjacency
// ---------------------------------------------------------------------------
__global__ void mask_deg_kernel(const int* __restrict__ adj,
                                float* __restrict__ maskf,
                                float* __restrict__ deg) {
    int j = blockIdx.x;
    int i = threadIdx.x;
    float m = (adj[j * NN + i] > 0) ? 1.0f : 0.0f;
    maskf[j * NN + i] = m;
    __shared__ float s[NN];
    s[i] = m;
    __syncthreads();
    for (int off = NN / 2; off > 0; off >>= 1) {
        if (i < off) s[i] += s[i + off];
        __syncthreads();
    }
    if (i == 0) deg[j] = s[0];
}

// ---------------------------------------------------------------------------
// r_agg[b,j,c] = sum_i mask[j,i] * edge_features[b,i,j,12+c]
// ---------------------------------------------------------------------------
__global__ void ragg_kernel(const float* __restrict__ maskf,
                            const float* __restrict__ ef,
                            float* __restrict__ ragg) {
    int bj = blockIdx.x;
    int b = bj >> 8;
    int j = bj & 255;
    int i = threadIdx.x;
    float m = maskf[j * NN + i];
    long eidx = (((long)(b * NN + i)) * NN + j) * EF + (EF - 3);
    __shared__ float s0[NN], s1[NN], s2[NN];
    s0[i] = m * ef[eidx + 0];
    s1[i] = m * ef[eidx + 1];
    s2[i] = m * ef[eidx + 2];
    __syncthreads();
    for (int off = NN / 2; off > 0; off >>= 1) {
        if (i < off) { s0[i] += s0[i + off]; s1[i] += s1[i + off]; s2[i] += s2[i + off]; }
        __syncthreads();
    }
    if (i == 0) {
        ragg[bj * 3 + 0] = s0[0];
        ragg[bj * 3 + 1] = s1[0];
        ragg[bj * 3 + 2] = s2[0];
    }
}

// ---------------------------------------------------------------------------
// emb[row,h] = relu(node_features[row,:] @ W_emb + b_emb)
// ---------------------------------------------------------------------------
__global__ void emb0_kernel(const float* __restrict__ nf,
                            const float* __restrict__ Wemb,
                            const float* __restrict__ bemb,
                            float* __restrict__ emb) {
    int row = blockIdx.x;
    int h = threadIdx.x;
    const float* x = nf + (long)row * NF;
    float acc = bemb[h];
    #pragma unroll
    for (int f = 0; f < NF; ++f) acc = fmaf(x[f], Wemb[f * HH + h], acc);
    emb[(long)row * HH + h] = fmaxf(acc, 0.0f);
}

// ---------------------------------------------------------------------------
// msg_base[row,h] = r_agg[row,:] @ Wr + deg[j] * msg_b[h]
// ---------------------------------------------------------------------------
__global__ void msgbase_kernel(const float* __restrict__ ragg,
                               const float* __restrict__ deg,
                               const float* __restrict__ Wr,
                               const float* __restrict__ mb,
                               float* __restrict__ msg) {
    int row = blockIdx.x;
    int h = threadIdx.x;
    int j = row & 255;
    const float* r = ragg + (long)row * 3;
    float acc = deg[j] * mb[h];
    acc = fmaf(r[0], Wr[0 * HH + h], acc);
    acc = fmaf(r[1], Wr[1 * HH + h], acc);
    acc = fmaf(r[2], Wr[2 * HH + h], acc);
    msg[(long)row * HH + h] = acc;
}

// ---------------------------------------------------------------------------
// Wave32 WMMA f32 GEMM, 16x32 output block per wave (two C tiles share one
// A fragment per K-step):  D = A @ B (+ Cin) (+ bias[col]) (relu?)
// Strides and K are compile-time so all K-loop loads use base + immediate
// offsets (24-bit IOFFSET) instead of per-iteration 64-bit address adds.
//   A: MxK row-major (LDA), B: KxN row-major (LDB), D/Cin: MxN (LDD)
//   grid: x = N/32, y = M/64 (4 waves per block), z = batch
// VGPR layouts per CDNA5 ISA 7.12.2:
//   A 16x4 : lane L -> M = L&15 ; element e -> K = e + 2*(L>>4)
//   B 4x16 : lane L -> N = L&15 ; element e -> K = e + 2*(L>>4)
//   C 16x16: lane L, vgpr r -> M = r + 8*(L>>4), N = L&15
// ---------------------------------------------------------------------------
template <int LDA, int LDB, int LDD, int K>
__global__ __launch_bounds__(128)
void wmma_gemm_f32(const float* __restrict__ A, long sA,
                   const float* __restrict__ B, long sB,
                   const float* __restrict__ Cin, long sC,
                   const float* __restrict__ bias,
                   float* __restrict__ D, long sD,
                   int relu) {
    const int wave = threadIdx.x >> 5;
    const int lane = threadIdx.x & 31;
    const int m0 = (blockIdx.y * 4 + wave) * 16;
    const int n0 = blockIdx.x * 32;
    const int bz = blockIdx.z;

    const float* Ab = A + (long)bz * sA;
    const float* Bb = B + (long)bz * sB;
    float*       Db = D + (long)bz * sD;

    const int mr = lane & 15;        // A row / B,C column within tile
    const int lh = lane >> 4;        // half-wave select
    const int kh = lh << 1;          // K sub-offset (0 or 2)
    const int nc0 = n0 + mr;         // column of tile 0
    const int nc1 = nc0 + 16;        // column of tile 1

    v8f c0, c1;
    if (Cin) {
        const float* Cp = Cin + (long)bz * sC;
        #pragma unroll
        for (int r = 0; r < 8; ++r) {
            long ro = (long)(m0 + r + 8 * lh) * LDD;
            c0[r] = Cp[ro + nc0];
            c1[r] = Cp[ro + nc1];
        }
    } else {
        #pragma unroll
        for (int r = 0; r < 8; ++r) { c0[r] = 0.0f; c1[r] = 0.0f; }
    }
    if (bias) {
        float bv0 = bias[nc0];
        float bv1 = bias[nc1];
        #pragma unroll
        for (int r = 0; r < 8; ++r) { c0[r] += bv0; c1[r] += bv1; }
    }

    const float* Arow  = Ab + (long)(m0 + mr) * LDA + kh;
    const float* Bcol0 = Bb + (long)kh * LDB + nc0;
    const float* Bcol1 = Bb + (long)kh * LDB + nc1;

    #pragma unroll 8
    for (int k = 0; k < K; k += 4) {
        // prefetch B a few K-steps ahead (strided cacheline walk)
        __builtin_prefetch(Bcol0 + (k + 32) * LDB, 0, 1);
        v2f a, bv0, bv1;
        a[0]   = Arow[k];
        a[1]   = Arow[k + 1];
        bv0[0] = Bcol0[k * LDB];
        bv0[1] = Bcol0[(k + 1) * LDB];
        bv1[0] = Bcol1[k * LDB];
        bv1[1] = Bcol1[(k + 1) * LDB];
        c0 = __builtin_amdgcn_wmma_f32_16x16x4_f32(false, a, false, bv0,
                                                   (short)0, c0, false, false);
        c1 = __builtin_amdgcn_wmma_f32_16x16x4_f32(false, a, false, bv1,
                                                   (short)0, c1, false, false);
    }

    if (relu) {
        #pragma unroll
        for (int r = 0; r < 8; ++r) {
            c0[r] = fmaxf(c0[r], 0.0f);
            c1[r] = fmaxf(c1[r], 0.0f);
        }
    }
    #pragma unroll
    for (int r = 0; r < 8; ++r) {
        long ro = (long)(m0 + r + 8 * lh) * LDD;
        Db[ro + nc0] = c0[r];
        Db[ro + nc1] = c1[r];
    }
}

// ---------------------------------------------------------------------------
// Final fused edge kernel: block per (b*N + i), thread per j.
// ---------------------------------------------------------------------------
__global__ __launch_bounds__(256)
void flows_kernel(const float* __restrict__ ef,
                  const float* __restrict__ tfeat,
                  const float* __restrict__ hi,
                  const float* __restrict__ hj,
                  const float* __restrict__ W1,
                  const float* __restrict__ b1,
                  const float* __restrict__ W2,
                  const float* __restrict__ b2,
                  float* __restrict__ out) {
    const int bi = blockIdx.x;           // b*N + i
    const int t = threadIdx.x;
    const int b = bi >> 8;

    __shared__ float s_hi[FHD];
    __shared__ float s_wr[3 * FHD];
    __shared__ float s_wt[TD * FHD];
    __shared__ float s_b1[FHD];
    __shared__ float s_w2[FHD];

    if (t < FHD) {
        s_hi[t] = hi[(long)bi * FHD + t];
        s_b1[t] = b1[t];
        s_w2[t] = W2[t];
    }
    for (int idx = t; idx < 3 * FHD; idx += 256)
        s_wr[idx] = W1[(2 * HH) * FHD + idx];          // rows 256..258 contiguous
    for (int idx = t; idx < TD * FHD; idx += 256)
        s_wt[idx] = W1[(2 * HH + 3) * FHD + idx];      // rows 259..262 contiguous
    __syncthreads();

    const int j = t;
    const long base = (long)bi * NN + j;
    const float* erp = ef + base * EF + (EF - 3);
    const float e0 = erp[0], e1 = erp[1], e2 = erp[2];
    const float* tp = tfeat + base * TD;
    const float t0 = tp[0], t1 = tp[1], t2 = tp[2], t3 = tp[3];
    const float* hjr = hj + (long)(b * NN + j) * FHD;

    float acc = 0.0f;
    #pragma unroll 8
    for (int f = 0; f < FHD; ++f) {
        float v = s_hi[f] + hjr[f] + s_b1[f];
        v = fmaf(e0, s_wr[f], v);
        v = fmaf(e1, s_wr[FHD + f], v);
        v = fmaf(e2, s_wr[2 * FHD + f], v);
        v = fmaf(t0, s_wt[f], v);
        v = fmaf(t1, s_wt[FHD + f], v);
        v = fmaf(t2, s_wt[2 * FHD + f], v);
        v = fmaf(t3, s_wt[3 * FHD + f], v);
        v = fmaxf(v, 0.0f);
        acc = fmaf(v, s_w2[f], acc);
    }
    out[base] = fmaxf(acc + b2[0], 0.0f);
}

// ---------------------------------------------------------------------------
extern "C" void kernel_launch(void* const* d_in, const int* in_sizes, int n_in,
                              void* d_out, int out_size, void* d_ws, size_t ws_size,
                              hipStream_t stream) {
    const float* node_features = (const float*)d_in[0];
    const float* edge_features = (const float*)d_in[1];
    const float* temporal      = (const float*)d_in[2];
    const int*   adjacency     = (const int*)d_in[3];
    const float* W_emb         = (const float*)d_in[4];
    const float* b_emb         = (const float*)d_in[5];
    const float* msg_W         = (const float*)d_in[6];
    const float* msg_b         = (const float*)d_in[7];
    const float* upd_W         = (const float*)d_in[8];
    const float* upd_b         = (const float*)d_in[9];
    const float* W1            = (const float*)d_in[10];
    const float* b1            = (const float*)d_in[11];
    const float* W2            = (const float*)d_in[12];
    const float* b2            = (const float*)d_in[13];
    float* out = (float*)d_out;

    // workspace layout (floats)
    float* w = (float*)d_ws;
    float* maskf = w;                        // 65536
    float* deg   = maskf + NN * NN;          // 256
    float* ragg  = deg + NN;                 // 6144
    float* embA  = ragg + NB * NN * 3;       // 262144
    float* embB  = embA + NB * NN * HH;      // 262144
    float* tmp   = embB + NB * NN * HH;      // 262144
    float* msg   = tmp  + NB * NN * HH;      // 262144
    float* hi    = msg  + NB * NN * HH;      // 131072
    float* hj    = hi   + NB * NN * FHD;     // 131072

    const int rows = NB * NN;                // 2048

    mask_deg_kernel<<<NN, NN, 0, stream>>>(adjacency, maskf, deg);
    ragg_kernel<<<NB * NN, NN, 0, stream>>>(maskf, edge_features, ragg);
    emb0_kernel<<<rows, HH, 0, stream>>>(node_features, W_emb, b_emb, embA);

    float* cur = embA;
    float* nxt = embB;
    for (int l = 0; l < LL; ++l) {
        const float* We   = msg_W + (long)l * (HH + 3) * HH;            // rows 0..H-1
        const float* Wr   = msg_W + ((long)l * (HH + 3) + HH) * HH;     // rows H..H+2
        const float* Utop = upd_W + (long)l * 2 * HH * HH;              // rows 0..H-1
        const float* Ubot = Utop + (long)HH * HH;                       // rows H..2H-1
        const float* mb   = msg_b + (long)l * HH;
        const float* ub   = upd_b + (long)l * HH;

        // tmp = emb @ We   (2048 x 128 x 128)
        wmma_gemm_f32<HH, HH, HH, HH><<<dim3(HH / 32, rows / 64, 1), 128, 0, stream>>>(
            cur, 0, We, 0, nullptr, 0, nullptr, tmp, 0, 0);

        // msg <- r_agg@Wr + deg*msg_b
        msgbase_kernel<<<rows, HH, 0, stream>>>(ragg, deg, Wr, mb, msg);

        // msg += mask @ tmp[b]   (batched 256 x 128 x 256, A shared)
        wmma_gemm_f32<NN, HH, HH, NN><<<dim3(HH / 32, NN / 64, NB), 128, 0, stream>>>(
            maskf, 0, tmp, (long)NN * HH,
            msg, (long)NN * HH, nullptr, msg, (long)NN * HH, 0);

        // nxt = emb @ Utop + upd_b
        wmma_gemm_f32<HH, HH, HH, HH><<<dim3(HH / 32, rows / 64, 1), 128, 0, stream>>>(
            cur, 0, Utop, 0, nullptr, 0, ub, nxt, 0, 0);

        // nxt = relu(nxt + msg @ Ubot)
        wmma_gemm_f32<HH, HH, HH, HH><<<dim3(HH / 32, rows / 64, 1), 128, 0, stream>>>(
            msg, 0, Ubot, 0, nxt, 0, nullptr, nxt, 0, 1);

        float* t2 = cur; cur = nxt; nxt = t2;
    }

    // hi = emb @ W1[0:H], hj = emb @ W1[H:2H]   (2048 x 64 x 128)
    wmma_gemm_f32<HH, FHD, FHD, HH><<<dim3(FHD / 32, rows / 64, 1), 128, 0, stream>>>(
        cur, 0, W1, 0, nullptr, 0, nullptr, hi, 0, 0);
    wmma_gemm_f32<HH, FHD, FHD, HH><<<dim3(FHD / 32, rows / 64, 1), 128, 0, stream>>>(
        cur, 0, W1 + (long)HH * FHD, 0, nullptr, 0, nullptr, hj, 0, 0);

    flows_kernel<<<NB * NN, NN, 0, stream>>>(
        edge_features, temporal, hi, hj, W1, b1, W2, b2, out);

    (void)in_sizes; (void)n_in; (void)out_size; (void)ws_size;
}